// PDMEncodeur_14216341750165
// MI455X (gfx1250) — compile-verified
//
#include <hip/hip_runtime.h>
#include <math.h>

typedef __attribute__((ext_vector_type(2))) float v2f;
typedef __attribute__((ext_vector_type(8))) float v8f;

#define LREF   32000
#define PDM    10
#define TPB    128              // t-values per block (8 waves x 16)
#define OUTPB  (TPB * PDM)      // 1280 outputs per block
#define BLOCK  256
#define WAVES  (BLOCK / 32)     // 8
#define SEG    (OUTPB / BLOCK)  // 5 outputs per thread in scan phase
#define KTABN  256              // 16 phases x 16 taps (zero padded)

// ---------------- coefficient table (computed ONCE in pass 0) ----------------
static __device__ __forceinline__ float kern_coef(int p, int k) {
  // torchaudio _get_sinc_resample_kernel, orig=1 new=10, hann, width=7,
  // rolloff 0.99, lowpass_filter_width=6. f64 math, cast f32 (matches ref).
  if (p >= PDM || k >= 15) return 0.0f;
  const double PI = 3.14159265358979323846264338327950288;
  double t = (-(double)p / 10.0 + (double)(k - 7)) * 0.99;
  t = fmin(6.0, fmax(-6.0, t));
  double w = cos(t * PI / 12.0);
  w = w * w;
  double tp = t * PI;
  double s = (tp == 0.0) ? 1.0 : sin(tp) / tp;
  return (float)(s * w * 0.99);
}

__global__ void pdm_pass0_coefs(float* __restrict__ ktab) {
  int i = threadIdx.x;            // 0..255
  int n = i >> 4, k = i & 15;     // ktab[n*16 + k]
  ktab[i] = kern_coef(n, k);
}

// ---------------- wave32 / block scan helpers (f64) ----------------
static __device__ __forceinline__ double wave_incl_scan(double v, int lane) {
#pragma unroll
  for (int d = 1; d < 32; d <<= 1) {
    double t = __shfl_up(v, d, 32);
    if (lane >= d) v += t;
  }
  return v;
}

// Exclusive block scan over BLOCK values; wsum must hold WAVES doubles.
// After return (and the internal barriers), wsum[WAVES-1] == block total.
static __device__ __forceinline__ double block_excl_scan(double v, int tid,
                                                         double* wsum) {
  const int lane = tid & 31, wave = tid >> 5;
  double incl = wave_incl_scan(v, lane);
  if (lane == 31) wsum[wave] = incl;
  __syncthreads();
  if (wave == 0) {
    double wv = (lane < WAVES) ? wsum[lane] : 0.0;
#pragma unroll
    for (int d = 1; d < WAVES; d <<= 1) {
      double t = __shfl_up(wv, d, 32);
      if (lane >= d) wv += t;
    }
    if (lane < WAVES) wsum[lane] = wv;  // inclusive scan of wave totals
  }
  __syncthreads();
  double waveOff = (wave == 0) ? 0.0 : wsum[wave - 1];
  return waveOff + incl - v;
}

// ---------------- WMMA conv tile ----------------
// One 16(t) x 16(phase) tile via 4 chained V_WMMA_F32_16X16X4_F32 (K=15 pad 16).
static __device__ __forceinline__ v8f conv_tile(const float* xs, int waveBase,
                                                int m, int hi, const v2f bm[4]) {
  v8f acc = {0.f, 0.f, 0.f, 0.f, 0.f, 0.f, 0.f, 0.f};
#pragma unroll
  for (int c = 0; c < 4; ++c) {
    int k0 = 4 * c + hi * 2;      // this lane's K pair within chunk c
    v2f a;
    a.x = xs[waveBase + m + k0];
    a.y = xs[waveBase + m + k0 + 1];
    acc = __builtin_amdgcn_wmma_f32_16x16x4_f32(
        false, a, false, bm[c], (short)0, acc, false, false);
  }
  return acc;
}

static __device__ __forceinline__ void load_bmat(const float* __restrict__ ktab,
                                                 int n, int hi, v2f bm[4]) {
#pragma unroll
  for (int c = 0; c < 4; ++c) {
    int k0 = 4 * c + hi * 2;                       // even -> 8B aligned
    bm[c] = *(const v2f*)(ktab + n * 16 + k0);
  }
}

static __device__ __forceinline__ void load_xtile(const float* __restrict__ xrow,
                                                  float* xs, int t0, int Lrow,
                                                  int tid) {
  // xs[j] = xpad[t0 + j], xpad[i] = x[i-7] zero-padded; j in [0,144)
  for (int j = tid; j < 144; j += BLOCK) {
    int g = t0 - 7 + j;
    xs[j] = (g >= 0 && g < Lrow) ? (0.5f * xrow[g] + 0.5f) : 0.0f;
  }
  // prefetch next block's tile into the cache hierarchy (global_prefetch_b8)
  if (tid == 0 && t0 + TPB < Lrow)
    __builtin_prefetch(xrow + t0 + TPB, 0, 0);
}

// ---------------- pass 1: conv + per-chunk f64 sums ----------------
__global__ void __launch_bounds__(BLOCK)
pdm_pass1_chunksum(const float* __restrict__ wav, const float* __restrict__ ktab,
                   double* __restrict__ chunkSum, int Lrow) {
  __shared__ float xs[144];
  __shared__ double wsum[WAVES];
  const int row = blockIdx.y, blk = blockIdx.x, tid = threadIdx.x;
  const int t0 = blk * TPB;
  load_xtile(wav + (size_t)row * Lrow, xs, t0, Lrow, tid);

  const int lane = tid & 31, wave = tid >> 5;
  const int m = lane & 15, hi = lane >> 4, n = m;  // n = phase column
  v2f bm[4];
  load_bmat(ktab, n, hi, bm);
  __syncthreads();
  v8f acc = conv_tile(xs, wave * 16, m, hi, bm);

  double s = 0.0;
  if (n < PDM) {
#pragma unroll
    for (int v = 0; v < 8; ++v) s += (double)acc[v];
  }
  (void)block_excl_scan(s, tid, wsum);   // leaves block total in wsum[WAVES-1]
  if (tid == 0) chunkSum[(size_t)row * gridDim.x + blk] = wsum[WAVES - 1];
}

// ---------------- pass 2: per-row exclusive scan of chunk sums ----------------
__global__ void __launch_bounds__(BLOCK)
pdm_pass2_rowscan(const double* __restrict__ chunkSum,
                  double* __restrict__ chunkPref, int nChunks) {
  __shared__ double wsum[WAVES];
  const int row = blockIdx.x, tid = threadIdx.x;
  const double* cs = chunkSum + (size_t)row * nChunks;
  double* cp = chunkPref + (size_t)row * nChunks;
  double carry = 0.0;
  for (int base = 0; base < nChunks; base += BLOCK) {
    int i = base + tid;
    double v = (i < nChunks) ? cs[i] : 0.0;
    double e = block_excl_scan(v, tid, wsum);
    if (i < nChunks) cp[i] = carry + e;
    carry += wsum[WAVES - 1];
    __syncthreads();
  }
}

// ---------------- pass 3: conv (recomputed) + scan + spikes ----------------
__global__ void __launch_bounds__(BLOCK)
pdm_pass3_spikes(const float* __restrict__ wav, const float* __restrict__ ktab,
                 const double* __restrict__ chunkPref, float* __restrict__ out,
                 int Lrow) {
  __shared__ float xs[144];
  __shared__ float ybuf[OUTPB];
  __shared__ double wsum[WAVES];
  const int row = blockIdx.y, blk = blockIdx.x, tid = threadIdx.x;
  const int t0 = blk * TPB;
  load_xtile(wav + (size_t)row * Lrow, xs, t0, Lrow, tid);

  const int lane = tid & 31, wave = tid >> 5;
  const int m = lane & 15, hi = lane >> 4, n = m;
  v2f bm[4];
  load_bmat(ktab, n, hi, bm);
  __syncthreads();
  v8f acc = conv_tile(xs, wave * 16, m, hi, bm);

  // Scatter D tile time-ordered: D row M = v + 8*hi -> t, col N = n -> phase
  if (n < PDM) {
#pragma unroll
    for (int v = 0; v < 8; ++v)
      ybuf[(wave * 16 + v + 8 * hi) * PDM + n] = acc[v];
  }
  __syncthreads();

  // 3-level scan: 5 elems/thread -> block scan of thread sums -> local sweep
  double s = 0.0;
#pragma unroll
  for (int i = 0; i < SEG; ++i) s += (double)ybuf[tid * SEG + i];
  double texcl = block_excl_scan(s, tid, wsum);

  double c = chunkPref[(size_t)row * gridDim.x + blk] + texcl;
  const bool firstGlobal = (blk == 0 && tid == 0);
  double dprev = firstGlobal ? -1.0 : floor(c);  // c == cumsum of all prior elems
  float* orow = out + (size_t)row * (size_t)Lrow * PDM + (size_t)t0 * PDM;
#pragma unroll
  for (int i = 0; i < SEG; ++i) {
    c += (double)ybuf[tid * SEG + i];
    double d = floor(c);
    orow[tid * SEG + i] = (d - dprev > 0.0) ? 1.0f : 0.0f;
    dprev = d;
  }
}

extern "C" void kernel_launch(void* const* d_in, const int* in_sizes, int n_in,
                              void* d_out, int out_size, void* d_ws, size_t ws_size,
                              hipStream_t stream) {
  (void)n_in; (void)out_size; (void)ws_size;
  const float* wav = (const float*)d_in[0];
  const int Lrow = LREF;
  const int B = in_sizes[0] / Lrow;        // 64
  const int chunks = Lrow / TPB;           // 250 chunks per row

  float* ktab = (float*)d_ws;                                   // 256 f32 (1 KB)
  double* chunkSum  = (double*)((char*)d_ws + KTABN * sizeof(float));
  double* chunkPref = chunkSum + (size_t)B * chunks;            // B*chunks f64 each

  dim3 grid(chunks, B);
  pdm_pass0_coefs<<<1, KTABN, 0, stream>>>(ktab);
  pdm_pass1_chunksum<<<grid, BLOCK, 0, stream>>>(wav, ktab, chunkSum, Lrow);
  pdm_pass2_rowscan<<<B, BLOCK, 0, stream>>>(chunkSum, chunkPref, chunks);
  pdm_pass3_spikes<<<grid, BLOCK, 0, stream>>>(wav, ktab, chunkPref,
                                               (float*)d_out, Lrow);
}